// ClipLoss_62878321213992
// MI455X (gfx1250) — compile-verified
//
#include <hip/hip_runtime.h>
#include <hip/hip_bf16.h>
#include <math.h>

#define B_SZ 1024
#define D_SZ 768

typedef __attribute__((ext_vector_type(2))) float v2f;
typedef __attribute__((ext_vector_type(8))) float v8f;

// ---------------------------------------------------------------------------
// GEMM: logits[m][n] = scale * sum_k img[m][k] * txt[n][k]
// V_WMMA_F32_16X16X4_F32, register-blocked: each wave owns a 16(M) x 64(N)
// strip = 4 accumulator tiles, so the A fragment is loaded once per K-step
// and reused for 4 WMMAs (1.25 loads/WMMA instead of 2).
// Block = 128 threads (4 waves) -> 16 x 256 strip. Grid = (64, 4).
// ---------------------------------------------------------------------------
__global__ __launch_bounds__(128) void gemm_logits(
    const float* __restrict__ img, const float* __restrict__ txt,
    const float* __restrict__ scale_p, float* __restrict__ logits) {
  const int lane  = threadIdx.x & 31;
  const int wave  = threadIdx.x >> 5;
  const int mBase = blockIdx.x * 16;
  const int nBase = blockIdx.y * 256 + wave * 64;
  const int l15   = lane & 15;
  const int kGrp  = (lane >> 4) * 2;   // lanes 0-15 -> K+{0,1}; lanes 16-31 -> K+{2,3}

  // A fragment: row = mBase+l15, cols k+kGrp..k+kGrp+1          (16x4 MxK fp32)
  // B fragment j: B[k][n] = txt[n][k]; N = nBase+16j+l15, same K cols
  const float* aPtr = img + (size_t)(mBase + l15) * D_SZ + kGrp;
  const float* bPtr = txt + (size_t)(nBase + l15) * D_SZ + kGrp;
  const size_t bStride = (size_t)16 * D_SZ;   // 48 KB: fits immediate offset

  v8f c0 = {}, c1 = {}, c2 = {}, c3 = {};
#pragma unroll 2
  for (int kk = 0; kk < D_SZ; kk += 4) {
    v2f a  = *(const v2f*)(aPtr + kk);
    v2f b0 = *(const v2f*)(bPtr + kk);
    v2f b1 = *(const v2f*)(bPtr + kk + bStride);
    v2f b2 = *(const v2f*)(bPtr + kk + 2 * bStride);
    v2f b3 = *(const v2f*)(bPtr + kk + 3 * bStride);
    // (neg_a, A, neg_b, B, c_mod, C, reuse_a, reuse_b)
    c0 = __builtin_amdgcn_wmma_f32_16x16x4_f32(false, a, false, b0, (short)0, c0, false, false);
    c1 = __builtin_amdgcn_wmma_f32_16x16x4_f32(false, a, false, b1, (short)0, c1, false, false);
    c2 = __builtin_amdgcn_wmma_f32_16x16x4_f32(false, a, false, b2, (short)0, c2, false, false);
    c3 = __builtin_amdgcn_wmma_f32_16x16x4_f32(false, a, false, b3, (short)0, c3, false, false);
  }

  const float s    = scale_p[0];
  const int   row0 = mBase + (lane >> 4) * 8;  // C/D: VGPR r -> M=r (lanes<16), M=r+8 (lanes>=16)
  float* outBase = logits + (size_t)row0 * B_SZ + nBase + l15;
#pragma unroll
  for (int r = 0; r < 8; ++r) {
    outBase[(size_t)r * B_SZ +  0] = c0[r] * s;
    outBase[(size_t)r * B_SZ + 16] = c1[r] * s;
    outBase[(size_t)r * B_SZ + 32] = c2[r] * s;
    outBase[(size_t)r * B_SZ + 48] = c3[r] * s;
  }
}

// ---------------------------------------------------------------------------
// Per-row 64-bit position-sensitive hash of the raw bit patterns.
// ---------------------------------------------------------------------------
__global__ __launch_bounds__(256) void row_hash(const float* __restrict__ img,
                                                unsigned long long* __restrict__ hash) {
  __shared__ unsigned long long sh[256];
  const int row = blockIdx.x;
  const unsigned* bits = (const unsigned*)(img + (size_t)row * D_SZ);
  unsigned long long h = 0ull;
  for (int k = threadIdx.x; k < D_SZ; k += 256) {
    unsigned long long v = (unsigned long long)bits[k];
    v = (v + 0x9E3779B97F4A7C15ull) * (unsigned long long)(2u * (unsigned)k + 1u);
    v ^= v >> 29;
    v *= 0xBF58476D1CE4E5B9ull;
    v ^= v >> 32;
    h += v;
  }
  sh[threadIdx.x] = h;
  __syncthreads();
  for (int s = 128; s > 0; s >>= 1) {
    if ((int)threadIdx.x < s) sh[threadIdx.x] += sh[threadIdx.x + s];
    __syncthreads();
  }
  if (threadIdx.x == 0) hash[row] = sh[0];
}

// ---------------------------------------------------------------------------
// labels[j] = first i <= j with row i == row j (float equality, ref semantics).
// Hash prefilter; exact verification only on hash hits (rare).
// ---------------------------------------------------------------------------
__global__ void labels_kernel(const float* __restrict__ img,
                              const unsigned long long* __restrict__ hash,
                              int* __restrict__ labels) {
  const int j = blockIdx.x * blockDim.x + threadIdx.x;
  if (j >= B_SZ) return;
  const unsigned long long hj = hash[j];
  int lab = j;
  for (int i = 0; i < j; ++i) {
    if (hash[i] == hj) {
      const float* ri = img + (size_t)i * D_SZ;
      const float* rj = img + (size_t)j * D_SZ;
      bool eq = true;
      for (int k = 0; k < D_SZ; ++k) {
        if (!(ri[k] == rj[k])) { eq = false; break; }
      }
      if (eq) { lab = i; break; }
    }
  }
  labels[j] = lab;
}

// ---------------------------------------------------------------------------
// Per-row loss: lse(row) - logits[row, label[row]].  One block per row.
// ---------------------------------------------------------------------------
__global__ __launch_bounds__(256) void loss_kernel(const float* __restrict__ logits,
                                                   const int* __restrict__ labels,
                                                   float* __restrict__ rowloss) {
  __shared__ float sh[256];
  const int row = blockIdx.x;
  const float* lrow = logits + (size_t)row * B_SZ;

  float m = -INFINITY;
  for (int k = threadIdx.x; k < B_SZ; k += 256) m = fmaxf(m, lrow[k]);
  sh[threadIdx.x] = m;
  __syncthreads();
  for (int s = 128; s > 0; s >>= 1) {
    if ((int)threadIdx.x < s) sh[threadIdx.x] = fmaxf(sh[threadIdx.x], sh[threadIdx.x + s]);
    __syncthreads();
  }
  m = sh[0];
  __syncthreads();

  float sum = 0.0f;
  for (int k = threadIdx.x; k < B_SZ; k += 256) sum += expf(lrow[k] - m);
  sh[threadIdx.x] = sum;
  __syncthreads();
  for (int s = 128; s > 0; s >>= 1) {
    if ((int)threadIdx.x < s) sh[threadIdx.x] += sh[threadIdx.x + s];
    __syncthreads();
  }
  if (threadIdx.x == 0) {
    const float lse = m + logf(sh[0]);
    rowloss[row] = lse - lrow[labels[row]];
  }
}

// Deterministic fixed-order final mean over 1024 per-row losses.
__global__ __launch_bounds__(256) void final_reduce(const float* __restrict__ rowloss,
                                                    float* __restrict__ out) {
  __shared__ float sh[256];
  float acc = 0.0f;
#pragma unroll
  for (int r = 0; r < 4; ++r) acc += rowloss[threadIdx.x + 256 * r];
  sh[threadIdx.x] = acc;
  __syncthreads();
  for (int s = 128; s > 0; s >>= 1) {
    if ((int)threadIdx.x < s) sh[threadIdx.x] += sh[threadIdx.x + s];
    __syncthreads();
  }
  if (threadIdx.x == 0) out[0] = sh[0] * (1.0f / (float)B_SZ);
}

extern "C" void kernel_launch(void* const* d_in, const int* in_sizes, int n_in,
                              void* d_out, int out_size, void* d_ws, size_t ws_size,
                              hipStream_t stream) {
  const float* img   = (const float*)d_in[0];
  const float* txt   = (const float*)d_in[1];
  const float* scale = (const float*)d_in[2];
  float* out = (float*)d_out;

  char* ws = (char*)d_ws;
  float* logits            = (float*)ws;                                                   // 4 MB
  unsigned long long* hash = (unsigned long long*)(ws + (size_t)B_SZ * B_SZ * 4);          // 8 KB
  int* labels              = (int*)(ws + (size_t)B_SZ * B_SZ * 4 + B_SZ * 8);              // 4 KB
  float* rowloss           = (float*)(ws + (size_t)B_SZ * B_SZ * 4 + B_SZ * 8 + B_SZ * 4); // 4 KB

  gemm_logits<<<dim3(B_SZ / 16, B_SZ / 256), 128, 0, stream>>>(img, txt, scale, logits);
  row_hash<<<B_SZ, 256, 0, stream>>>(img, hash);
  labels_kernel<<<B_SZ / 256, 256, 0, stream>>>(img, hash, labels);
  loss_kernel<<<B_SZ, 256, 0, stream>>>(logits, labels, rowloss);
  final_reduce<<<1, 256, 0, stream>>>(rowloss, out);
}